// LIFMatchedBank_17059610100023
// MI455X (gfx1250) — compile-verified
//
#include <hip/hip_runtime.h>

typedef __attribute__((ext_vector_type(2))) float v2f;
typedef __attribute__((ext_vector_type(8))) float v8f;

#define B_TOT 256
#define DIN   128
#define T_TOT 2000
#define K_TOT 64
#define WLS   132                      // LDS W row stride (floats): 4n banks -> conflict-free b64
#define BKT   ((size_t)B_TOT * K_TOT * T_TOT)

// ---------------------------------------------------------------------------
// Pass 1: I[b,k,t] = sum_d W[k,d] * x[b,d,t], written into the V_hist half of
// d_out (used as scratch; pass 2 overwrites it in place with V).
// One wave per (b, 16-wide t-tile); x fragments held in 64 VGPRs and reused
// across all 4 k-tiles; W read from padded LDS (conflict-free ds_load_b64).
// Grid: (256 b) x (25);  block: 160 threads = 5 waves -> 125 t-tiles per b.
// ---------------------------------------------------------------------------
__global__ __launch_bounds__(160, 1)
void lif_gemm_wmma_kernel(const float* __restrict__ x,
                          const float* __restrict__ W,
                          float* __restrict__ out)
{
    __shared__ float Wl[K_TOT * WLS];
    for (int i = threadIdx.x; i < K_TOT * DIN; i += 160) {
        const int k = i >> 7;
        const int d = i & 127;
        Wl[k * WLS + d] = W[i];
    }
    __syncthreads();

    const int lane = threadIdx.x & 31;
    const int wave = threadIdx.x >> 5;          // 0..4
    const int n    = lane & 15;                 // t column (B/D N index, A row M)
    const int kh   = lane >> 4;                 // K-pair half select
    const int b    = blockIdx.x;
    const int t0   = (blockIdx.y * 5 + wave) * 16;

    // B-matrix fragments: bf[s] = { x[b][4s+2kh][t0+n], x[b][4s+2kh+1][t0+n] }
    const float* xb = x + (size_t)b * DIN * T_TOT + t0 + n;
    v2f bf[32];
#pragma unroll
    for (int s = 0; s < 32; ++s) {
        const float* p = xb + (size_t)(4 * s + 2 * kh) * T_TOT;
        bf[s].x = p[0];
        bf[s].y = p[T_TOT];
    }

    float* outI = out + BKT + (size_t)b * K_TOT * T_TOT + t0 + n;

#pragma unroll
    for (int kt = 0; kt < 4; ++kt) {
        const float* wl = Wl + (kt * 16 + n) * WLS + 2 * kh;
        v8f c0 = {0.f, 0.f, 0.f, 0.f, 0.f, 0.f, 0.f, 0.f};
        v8f c1 = {0.f, 0.f, 0.f, 0.f, 0.f, 0.f, 0.f, 0.f};
#pragma unroll
        for (int s = 0; s < 32; s += 2) {
            v2f a0 = *(const v2f*)(wl + 4 * s);
            c0 = __builtin_amdgcn_wmma_f32_16x16x4_f32(false, a0, false, bf[s],
                                                       (short)0, c0, false, false);
            v2f a1 = *(const v2f*)(wl + 4 * (s + 1));
            c1 = __builtin_amdgcn_wmma_f32_16x16x4_f32(false, a1, false, bf[s + 1],
                                                       (short)0, c1, false, false);
        }
        // C/D layout: lane n, row M = kh*8 + i -> k = kt*16 + M; coalesced b32 stores
#pragma unroll
        for (int i = 0; i < 8; ++i) {
            const int m = kh * 8 + i;
            outI[(size_t)(kt * 16 + m) * T_TOT] = c0[i] + c1[i];
        }
    }
}

// ---------------------------------------------------------------------------
// Pass 2: sequential LIF scan. One thread per (b,k) chain (16384 chains).
// Reads I from the V_hist half of d_out in 64B-aligned 16-step chunks,
// overwrites it in place with V, writes S to the first half. Bias folded in.
// Grid: 256 blocks x 64 threads.
// ---------------------------------------------------------------------------
__global__ __launch_bounds__(64, 1)
void lif_scan_kernel(const float* __restrict__ bias,
                     const float* __restrict__ vth,
                     const float* __restrict__ tau,
                     float* __restrict__ out)
{
    const int c = blockIdx.x * 64 + threadIdx.x;     // chain id = b*64 + k
    const int k = c & 63;
    const float biasv  = bias[k];
    const float vthv   = vth[k];
    const float alphav = __expf(-1.0f / tau[k]);

    float* pS = out + (size_t)c * T_TOT;             // S_hist
    float* pI = out + BKT + (size_t)c * T_TOT;       // I in, V out (in place)

    float v = 0.0f;
    for (int t0 = 0; t0 < T_TOT; t0 += 16) {
        float a[16];
#pragma unroll
        for (int q = 0; q < 4; ++q) {
            const float4 i4 = *(const float4*)(pI + t0 + 4 * q);
            a[4 * q + 0] = i4.x; a[4 * q + 1] = i4.y;
            a[4 * q + 2] = i4.z; a[4 * q + 3] = i4.w;
        }
        float s[16];
#pragma unroll
        for (int j = 0; j < 16; ++j) {
            v = alphav * v + a[j] + biasv;
            const float sp = (v > vthv) ? 1.0f : 0.0f;
            v -= sp * vthv;
            s[j] = sp;
            a[j] = v;
        }
#pragma unroll
        for (int q = 0; q < 4; ++q) {
            *(float4*)(pS + t0 + 4 * q) =
                make_float4(s[4 * q + 0], s[4 * q + 1], s[4 * q + 2], s[4 * q + 3]);
            *(float4*)(pI + t0 + 4 * q) =
                make_float4(a[4 * q + 0], a[4 * q + 1], a[4 * q + 2], a[4 * q + 3]);
        }
    }
}

extern "C" void kernel_launch(void* const* d_in, const int* in_sizes, int n_in,
                              void* d_out, int out_size, void* d_ws, size_t ws_size,
                              hipStream_t stream) {
    (void)in_sizes; (void)n_in; (void)out_size; (void)d_ws; (void)ws_size;
    const float* x    = (const float*)d_in[0];
    const float* W    = (const float*)d_in[1];
    const float* bias = (const float*)d_in[2];
    const float* vth  = (const float*)d_in[3];
    const float* tau  = (const float*)d_in[4];
    float* out = (float*)d_out;

    // Pass 1: 256 x 25 blocks, 160 threads (5 waves) -> 32000 WMMA wave-tiles
    lif_gemm_wmma_kernel<<<dim3(B_TOT, 25), dim3(160), 0, stream>>>(x, W, out);
    // Pass 2: 16384 independent chains
    lif_scan_kernel<<<dim3(256), dim3(64), 0, stream>>>(bias, vth, tau, out);
}